// MambaVision3DMixer_36859409334927
// MI455X (gfx1250) — compile-verified
//
#include <hip/hip_runtime.h>
#include <hip/hip_bf16.h>

#define Bn 2
#define Cn 96
#define HALFC 48
#define Dd 64
#define Hh 64
#define Ww 64
#define Vv (Dd*Hh*Ww)          /* 262144 voxels per (b,c) */
#define EPSf 1e-05f

typedef __attribute__((ext_vector_type(2))) float v2f;
typedef __attribute__((ext_vector_type(8))) float v8f;

// ---------------------------------------------------------------------------
// Kernel A: per-(b,c) mean/var over D*H*W, folded with gamma/beta into
// scale/shift so the GEMM kernel can normalize with a single FMA per element.
// ---------------------------------------------------------------------------
__global__ void k_stats(const float* __restrict__ x,
                        const float* __restrict__ gamma,
                        const float* __restrict__ beta,
                        float* __restrict__ scale,
                        float* __restrict__ shift) {
  const int c = blockIdx.x;
  const int b = blockIdx.y;
  const float* xc = x + (size_t)(b * Cn + c) * Vv;
  const int tid = threadIdx.x;

  float s = 0.f, ss = 0.f;
  for (int i = tid * 4; i < Vv; i += 256 * 4) {
    float4 v = *reinterpret_cast<const float4*>(xc + i);
    s  += v.x + v.y + v.z + v.w;
    ss += v.x*v.x + v.y*v.y + v.z*v.z + v.w*v.w;
  }
  __shared__ float rs[256];
  __shared__ float rss[256];
  rs[tid] = s; rss[tid] = ss;
  __syncthreads();
  for (int off = 128; off > 0; off >>= 1) {
    if (tid < off) { rs[tid] += rs[tid + off]; rss[tid] += rss[tid + off]; }
    __syncthreads();
  }
  if (tid == 0) {
    const float inv = 1.f / (float)Vv;
    float mean = rs[0] * inv;
    float var  = rss[0] * inv - mean * mean;
    float rstd = rsqrtf(var + EPSf);
    float sc = rstd * gamma[c];
    scale[b * Cn + c] = sc;
    shift[b * Cn + c] = beta[c] - mean * sc;
  }
}

// ---------------------------------------------------------------------------
// Channel GEMM via V_WMMA_F32_16X16X4_F32:  dst[o][v] = sum_c W[o][c]*src[c][v]
//   + bias, with optional inline normalization of src (NORM) and optional
//   residual add (RES).  One wave owns a 16-voxel N-tile and the full M=96
//   (6 v8f accumulators); K=96 is consumed in 24 steps of 4.
// A-fragment (16x4 f32, 2 VGPRs): lanes 0-15 -> M, lanes 16-31 -> M with
//   K offset +2; VGPR0 = K {0|2}, VGPR1 = K {1|3}.  B mirrors with N on lanes.
// C/D (v8f): VGPR j = row M=j (lanes 0-15) / M=j+8 (lanes 16-31), N = lane&15.
// ---------------------------------------------------------------------------
#define GWAVES 4
#define GCHUNK 8

template<bool NORM, bool RES>
__global__ void k_chan_gemm(const float* __restrict__ src,    // [B][C][V]
                            const float* __restrict__ wmat,   // [C][C] row-major (o,c)
                            const float* __restrict__ bias,   // [C]
                            const float* __restrict__ scale,  // [B*C] (NORM only)
                            const float* __restrict__ shift,  // [B*C] (NORM only)
                            const float* __restrict__ resid,  // [B][C][V] (RES only)
                            float* __restrict__ dst) {        // [B][C][V]
  __shared__ float s_w[Cn * Cn];
  __shared__ float s_scale[Cn];
  __shared__ float s_shift[Cn];
  __shared__ float s_bias[Cn];

  const int tid = threadIdx.x;
  const int b = blockIdx.y;
  for (int i = tid; i < Cn * Cn; i += 128) s_w[i] = wmat[i];
  if (tid < Cn) {
    s_bias[tid] = bias[tid];
    if (NORM) {
      s_scale[tid] = scale[b * Cn + tid];
      s_shift[tid] = shift[b * Cn + tid];
    }
  }
  __syncthreads();

  const int wave  = tid >> 5;
  const int lane  = tid & 31;
  const int nl    = lane & 15;
  const int khalf = (lane >> 4) << 1;   // 0 | 2 : K offset within step
  const int mrow  = (lane >> 4) << 3;   // 0 | 8 : M offset in C/D rows

  const float* sb = src + (size_t)b * Cn * Vv;
  const float* rb = RES ? (resid + (size_t)b * Cn * Vv) : nullptr;
  float*       db = dst + (size_t)b * Cn * Vv;

  const int chunk0 = (blockIdx.x * GWAVES + wave) * GCHUNK;
  for (int ch = 0; ch < GCHUNK; ++ch) {
    const int v0 = (chunk0 + ch) * 16;

    v8f acc[6];
#pragma unroll
    for (int mt = 0; mt < 6; ++mt)
      acc[mt] = (v8f){0.f,0.f,0.f,0.f,0.f,0.f,0.f,0.f};

#pragma unroll 4
    for (int kk = 0; kk < Cn / 4; ++kk) {
      const int c0 = kk * 4 + khalf;
      float b0 = sb[(size_t)c0 * Vv + v0 + nl];
      float b1 = sb[(size_t)(c0 + 1) * Vv + v0 + nl];
      if (NORM) {
        b0 = b0 * s_scale[c0]     + s_shift[c0];
        b1 = b1 * s_scale[c0 + 1] + s_shift[c0 + 1];
      }
      const v2f Bf = {b0, b1};
#pragma unroll
      for (int mt = 0; mt < 6; ++mt) {
        const v2f Af = { s_w[(mt * 16 + nl) * Cn + c0],
                         s_w[(mt * 16 + nl) * Cn + c0 + 1] };
        acc[mt] = __builtin_amdgcn_wmma_f32_16x16x4_f32(
            /*neg_a=*/false, Af, /*neg_b=*/false, Bf,
            /*c_mod=*/(short)0, acc[mt], /*reuse_a=*/false, /*reuse_b=*/false);
      }
    }

#pragma unroll
    for (int mt = 0; mt < 6; ++mt) {
#pragma unroll
      for (int j = 0; j < 8; ++j) {
        const int m = mt * 16 + mrow + j;
        const size_t idx = (size_t)m * Vv + v0 + nl;
        float r = acc[mt][j] + s_bias[m];
        if (RES) r += rb[idx];
        db[idx] = r;
      }
    }
  }
}

// ---------------------------------------------------------------------------
// Depthwise 3x3x3 chains on a 4x8x32 spatial tile with halo-2 in LDS.
//   c <  48 : y = dw(silu(dw(xp, w_ssm)+b_ssm), w_mamba)+b_mamba
//   c >= 48 : y = silu(dw(xp, w_sym)+b_sym)
// Depthwise weights are block-uniform -> scalar loads / SGPR-resident.
// ---------------------------------------------------------------------------
#define TDm 4
#define THm 8
#define TWm 32
#define SXD (TDm + 4)
#define SXH (THm + 4)
#define SXW (TWm + 4)
#define STD (TDm + 2)
#define STH (THm + 2)
#define STW (TWm + 2)

__device__ __forceinline__ float silu_f(float v) {
  return v * (1.f / (1.f + __expf(-v)));
}

__global__ void k_dwchain(const float* __restrict__ xp,
                          const float* __restrict__ w_ssm, const float* __restrict__ b_ssm,
                          const float* __restrict__ w_mam, const float* __restrict__ b_mam,
                          const float* __restrict__ w_sym, const float* __restrict__ b_sym,
                          float* __restrict__ y) {
  __shared__ float sx[SXD * SXH * SXW];   // xp tile + halo 2
  __shared__ float st[STD * STH * STW];   // intermediate (branch 1)

  const int tid = threadIdx.x;
  const int bc = blockIdx.y;              // b*Cn + c
  const int c = bc % Cn;
  const int t = blockIdx.x;
  const int tw = t & 1, th = (t >> 1) & 7, td = t >> 4;
  const int d0 = td * TDm, h0 = th * THm, w0 = tw * TWm;
  const size_t base = (size_t)bc * Vv;

  for (int i = tid; i < SXD * SXH * SXW; i += 256) {
    const int lw = i % SXW;
    const int r1 = i / SXW;
    const int lh = r1 % SXH;
    const int ld = r1 / SXH;
    const int gd = d0 + ld - 2, gh = h0 + lh - 2, gw = w0 + lw - 2;
    float v = 0.f;
    if ((unsigned)gd < (unsigned)Dd && (unsigned)gh < (unsigned)Hh &&
        (unsigned)gw < (unsigned)Ww)
      v = xp[base + ((size_t)gd * Hh + gh) * Ww + gw];
    sx[i] = v;
  }
  __syncthreads();

  if (c < HALFC) {
    float w1[27], w2[27];
#pragma unroll
    for (int k = 0; k < 27; ++k) { w1[k] = w_ssm[c * 27 + k]; w2[k] = w_mam[c * 27 + k]; }
    const float bb1 = b_ssm[c], bb2 = b_mam[c];

    // intermediate t = silu(dw(xp)) on interior + halo 1
    for (int i = tid; i < STD * STH * STW; i += 256) {
      const int jw = i % STW;
      const int r1 = i / STW;
      const int jh = r1 % STH;
      const int jd = r1 / STH;
      float acc = bb1;
#pragma unroll
      for (int kd = 0; kd < 3; ++kd)
#pragma unroll
        for (int kh = 0; kh < 3; ++kh)
#pragma unroll
          for (int kw = 0; kw < 3; ++kw)
            acc += w1[(kd * 3 + kh) * 3 + kw] *
                   sx[((jd + kd) * SXH + (jh + kh)) * SXW + (jw + kw)];
      st[i] = silu_f(acc);
    }
    __syncthreads();

    for (int i = tid; i < TDm * THm * TWm; i += 256) {
      const int ow = i % TWm;
      const int r1 = i / TWm;
      const int oh = r1 % THm;
      const int od = r1 / THm;
      float acc = bb2;
#pragma unroll
      for (int kd = 0; kd < 3; ++kd)
#pragma unroll
        for (int kh = 0; kh < 3; ++kh)
#pragma unroll
          for (int kw = 0; kw < 3; ++kw)
            acc += w2[(kd * 3 + kh) * 3 + kw] *
                   st[((od + kd) * STH + (oh + kh)) * STW + (ow + kw)];
      y[base + ((size_t)(d0 + od) * Hh + (h0 + oh)) * Ww + (w0 + ow)] = acc;
    }
  } else {
    float w1[27];
#pragma unroll
    for (int k = 0; k < 27; ++k) w1[k] = w_sym[(c - HALFC) * 27 + k];
    const float bb = b_sym[c - HALFC];

    for (int i = tid; i < TDm * THm * TWm; i += 256) {
      const int ow = i % TWm;
      const int r1 = i / TWm;
      const int oh = r1 % THm;
      const int od = r1 / THm;
      float acc = bb;
#pragma unroll
      for (int kd = 0; kd < 3; ++kd)
#pragma unroll
        for (int kh = 0; kh < 3; ++kh)
#pragma unroll
          for (int kw = 0; kw < 3; ++kw)
            acc += w1[(kd * 3 + kh) * 3 + kw] *
                   sx[((od + kd + 1) * SXH + (oh + kh + 1)) * SXW + (ow + kw + 1)];
      y[base + ((size_t)(d0 + od) * Hh + (h0 + oh)) * Ww + (w0 + ow)] = silu_f(acc);
    }
  }
}

// ---------------------------------------------------------------------------
extern "C" void kernel_launch(void* const* d_in, const int* in_sizes, int n_in,
                              void* d_out, int out_size, void* d_ws, size_t ws_size,
                              hipStream_t stream) {
  const float* x      = (const float*)d_in[0];
  const float* gamma  = (const float*)d_in[1];
  const float* beta   = (const float*)d_in[2];
  const float* in_w   = (const float*)d_in[3];
  const float* in_b   = (const float*)d_in[4];
  const float* w_ssm  = (const float*)d_in[5];
  const float* b_ssm  = (const float*)d_in[6];
  const float* w_mam  = (const float*)d_in[7];
  const float* b_mam  = (const float*)d_in[8];
  const float* w_sym  = (const float*)d_in[9];
  const float* b_sym  = (const float*)d_in[10];
  const float* out_w  = (const float*)d_in[11];
  const float* out_b  = (const float*)d_in[12];
  float* out = (float*)d_out;

  float* ws    = (float*)d_ws;
  float* scale = ws;                  // 192 floats
  float* shift = ws + 256;            // 192 floats
  float* xp    = ws + 1024;           // B*C*V floats
  float* yv    = xp + (size_t)Bn * Cn * Vv;

  // 1) stats -> scale/shift
  k_stats<<<dim3(Cn, Bn), 256, 0, stream>>>(x, gamma, beta, scale, shift);

  // 2) fused norm + in-projection GEMM (WMMA f32 16x16x4)
  const int gx = Vv / (16 * GWAVES * GCHUNK);   // 512
  k_chan_gemm<true, false><<<dim3(gx, Bn), 128, 0, stream>>>(
      x, in_w, in_b, scale, shift, nullptr, xp);

  // 3) depthwise chains
  k_dwchain<<<dim3((Dd / TDm) * (Hh / THm) * (Ww / TWm), Bn * Cn), 256, 0, stream>>>(
      xp, w_ssm, b_ssm, w_mam, b_mam, w_sym, b_sym, yv);

  // 4) out-projection GEMM + bias + residual (WMMA f32 16x16x4)
  k_chan_gemm<false, true><<<dim3(gx, Bn), 128, 0, stream>>>(
      yv, out_w, out_b, nullptr, nullptr, x, out);
}